// Clustering_60979945668666
// MI455X (gfx1250) — compile-verified
//
#include <hip/hip_runtime.h>
#include <stdint.h>

// out[b,l,j] = x[b,l,order[j]], order = stable argsort(y_pred).
// Pure bandwidth problem: 256MB in + 256MB out @ 23.3 TB/s => ~22us floor.
// Strategy: both HBM directions fully coalesced b128; permutation resolved in LDS.
// Uses gfx1250 async global->LDS DMA path (ASYNCcnt) for the staging copies.
// Tile stream is non-temporal (one-touch); order table stays RT (L2-resident,
// re-read by every block).

typedef __attribute__((ext_vector_type(4))) float f4;

#define C               512                       // channels (innermost, contiguous)
#define ROWS_PER_BLOCK  8                         // 8 rows * 2KB = 16KB contiguous tile
#define THREADS         256
#define TILE_F4         (ROWS_PER_BLOCK * C / 4)  // 1024 float4 per tile
#define F4_PER_THREAD   (TILE_F4 / THREADS)       // 4

// ---------- gfx1250 async global->LDS helpers (inline asm, ASYNCcnt-tracked) ----------
__device__ __forceinline__ void async_g2l_b128_nt(unsigned lds_off, unsigned voff, const void* sbase) {
    // GVS form: global address is sbase plus the 32-bit offset in voff; LDS byte offset comes from lds_off
    // NT: one-touch stream, evict early from caches.
    asm volatile("global_load_async_to_lds_b128 %0, %1, %2 th:TH_LOAD_NT"
                 :: "v"(lds_off), "v"(voff), "s"(sbase) : "memory");
}
__device__ __forceinline__ void async_g2l_b64(unsigned lds_off, unsigned voff, const void* sbase) {
    // RT policy: order table is re-read by all blocks; keep it L2-resident.
    asm volatile("global_load_async_to_lds_b64 %0, %1, %2"
                 :: "v"(lds_off), "v"(voff), "s"(sbase) : "memory");
}
__device__ __forceinline__ void wait_async0() {
    asm volatile("s_wait_asynccnt 0" ::: "memory");
}

// ---------- kernel 1: stable argsort of y_pred (512 labels) -> order[512] ----------
// pos(c) = #{c' : y[c'] < y[c]} + #{c' < c : y[c'] == y[c]}  (stable rank)
// order[pos(c)] = c  is the inverse permutation == stable argsort.
__global__ void build_order_kernel(const int* __restrict__ y, int* __restrict__ order) {
    __shared__ int sy[C];
    const int t = threadIdx.x;          // one block of C threads
    sy[t] = y[t];
    __syncthreads();
    const int lbl = sy[t];
    int pos = 0;
#pragma unroll 8
    for (int c = 0; c < C; ++c) {
        const int lc = sy[c];
        pos += ((lc < lbl) | ((lc == lbl) & (c < t))) ? 1 : 0;
    }
    order[pos] = t;
}

// ---------- kernel 2: streaming permute through LDS ----------
__global__ void permute_kernel(const float* __restrict__ x,
                               const int*   __restrict__ order_g,
                               float*       __restrict__ out) {
    __shared__ __align__(16) float tile[ROWS_PER_BLOCK * C];   // 16 KB
    __shared__ __align__(16) int   s_order[C];                 // 2 KB

    const int    t       = threadIdx.x;
    const size_t block_f = (size_t)blockIdx.x * (ROWS_PER_BLOCK * C); // tile start (floats)

    // Async-stage the permutation table: 512 ints, 8 bytes/thread (RT, L2-hot).
    {
        const unsigned lds_base = (unsigned)(uintptr_t)(void*)s_order;
        const unsigned voff     = (unsigned)(t * 8);
        async_g2l_b64(lds_base + voff, voff, (const void*)order_g);
    }
    // Async-stage the 16KB contiguous tile: 4 x b128 per thread, coalesced, NT.
    {
        const unsigned lds_base = (unsigned)(uintptr_t)(void*)tile;
        const void*    gbase    = (const void*)(x + block_f);
#pragma unroll
        for (int k = 0; k < F4_PER_THREAD; ++k) {
            const unsigned voff = (unsigned)((t + k * THREADS) * 16);
            async_g2l_b128_nt(lds_base + voff, voff, gbase);
        }
    }
    wait_async0();       // this wave's LDS writes landed
    __syncthreads();     // all waves' writes visible

    // Gather inside LDS, store coalesced non-temporal b128 (one-touch stream).
    f4* __restrict__ out4 = (f4*)(out + block_f);
#pragma unroll
    for (int k = 0; k < F4_PER_THREAD; ++k) {
        const int i = t + k * THREADS;   // float4 index within tile [0, 1024)
        const int r = i >> 7;            // row within tile (C/4 = 128 groups/row)
        const int g = i & 127;           // output channel group
        const int j0 = s_order[4 * g + 0];
        const int j1 = s_order[4 * g + 1];
        const int j2 = s_order[4 * g + 2];
        const int j3 = s_order[4 * g + 3];
        const float* rowp = &tile[r * C];
        f4 v;
        v.x = rowp[j0];
        v.y = rowp[j1];
        v.z = rowp[j2];
        v.w = rowp[j3];
        __builtin_nontemporal_store(v, &out4[i]);
    }
}

extern "C" void kernel_launch(void* const* d_in, const int* in_sizes, int n_in,
                              void* d_out, int out_size, void* d_ws, size_t ws_size,
                              hipStream_t stream) {
    const float* x   = (const float*)d_in[0];   // [32, 4096, 512] fp32, C contiguous
    const int*   y   = (const int*)d_in[1];     // [512] int32 cluster ids
    float*       out = (float*)d_out;           // [32, 4096, 512] fp32
    int*         order = (int*)d_ws;            // 512 ints of scratch

    // 1) stable argsort of y_pred -> order (one tiny block)
    build_order_kernel<<<1, C, 0, stream>>>(y, order);

    // 2) streaming channel permutation (rows = B*L = in_sizes[0]/C, divisible by 8)
    const int nrows   = in_sizes[0] / C;              // 131072
    const int nblocks = nrows / ROWS_PER_BLOCK;       // 16384
    permute_kernel<<<nblocks, THREADS, 0, stream>>>(x, order, out);
}